// GCN_49323404427971
// MI455X (gfx1250) — compile-verified
//
#include <hip/hip_runtime.h>
#include <hip/hip_bf16.h>

typedef __attribute__((ext_vector_type(2))) float v2f;
typedef __attribute__((ext_vector_type(8))) float v8f;

#define IN_C 128
#define HIDC 64
#define OUTC 40
#define OUTP 48   // OUTC padded to a multiple of 16 for WMMA column tiles

// ---------------- small utility kernels ----------------

__global__ void fill_f32(float* __restrict__ p, float v, int n) {
    int i = blockIdx.x * blockDim.x + threadIdx.x;
    if (i < n) p[i] = v;
}

__global__ void deg_accum(float* __restrict__ deg, const int* __restrict__ dst,
                          const float* __restrict__ w, int E) {
    int e = blockIdx.x * blockDim.x + threadIdx.x;
    if (e < E) atomicAdd(&deg[dst[e]], w[e]);
}

__global__ void rsqrt_inplace(float* __restrict__ d, int n) {
    int i = blockIdx.x * blockDim.x + threadIdx.x;
    if (i < n) {
        float v = d[i];
        d[i] = (v > 0.0f) ? rsqrtf(v) : 0.0f;
    }
}

__global__ void compute_norm(float* __restrict__ norm, const int* __restrict__ src,
                             const int* __restrict__ dst, const float* __restrict__ w,
                             const float* __restrict__ dinv, int E) {
    int e = blockIdx.x * blockDim.x + threadIdx.x;
    if (e < E) norm[e] = dinv[src[e]] * w[e] * dinv[dst[e]];
}

// zero-pad W3 (64x40) into 64x48 so the WMMA B-loads need no guards
__global__ void pad_w3(const float* __restrict__ W, float* __restrict__ Wp) {
    int i = blockIdx.x * blockDim.x + threadIdx.x;
    if (i >= HIDC * OUTP) return;
    int r = i / OUTP, c = i - r * OUTP;
    Wp[i] = (c < OUTC) ? W[r * OUTC + c] : 0.0f;
}

// ---------------- fp32 WMMA GEMM: H[M,NCS] = X[M,K] @ W[K,NC] ----------------
// Compile-time K/NC: no runtime guards in the hot loop, constant strides,
// unrolled so fragment loads overlap v_wmma_f32_16x16x4_f32 execution.
// One wave per 16x16 output tile.
// A layout: lanes 0-15 hold row M=lane, K = k0+{0,1}; lanes 16-31 hold K = k0+{2,3}.
// B layout: lane&15 = column N; lane>>4 selects K-pair, mirrored across 2 VGPRs.
// C/D layout: VGPR v, lanes 0-15 -> M=v, lanes 16-31 -> M=v+8; N = lane&15.

template <int K, int NC, int NCS>
__global__ void gemm_wmma_f32(const float* __restrict__ X, const float* __restrict__ W,
                              float* __restrict__ H, int M) {
    constexpr int TILES_N = NC / 16;
    const int gwave = (blockIdx.x * blockDim.x + threadIdx.x) >> 5;
    const int lane  = threadIdx.x & 31;
    const int tm = gwave / TILES_N;
    const int tn = gwave - tm * TILES_N;
    if (tm * 16 >= M) return;                       // uniform per-wave exit

    const int row   = tm * 16 + (lane & 15);
    const int colB  = tn * 16 + (lane & 15);
    const int khalf = (lane >> 4) << 1;             // 0 or 2

    const float* Xr = X + (size_t)row * K + khalf;  // A: two consecutive floats / step
    const float* Wc = W + (size_t)khalf * NC + colB;

    v8f c = {};
#pragma unroll 8
    for (int k0 = 0; k0 < K; k0 += 4) {
        v2f a, b;
        a.x = Xr[k0];
        a.y = Xr[k0 + 1];
        b.x = Wc[(size_t)k0 * NC];
        b.y = Wc[(size_t)k0 * NC + NC];
        c = __builtin_amdgcn_wmma_f32_16x16x4_f32(
                /*neg_a=*/false, a, /*neg_b=*/false, b,
                /*c_mod=*/(short)0, c, /*reuse_a=*/false, /*reuse_b=*/false);
    }

    const int crow = tm * 16 + ((lane >> 4) << 3);  // +8 for upper half-wave
    const int ccol = tn * 16 + (lane & 15);
    if (ccol < NCS) {                               // folds away when NCS == NC
#pragma unroll
        for (int v = 0; v < 8; ++v)
            H[(size_t)(crow + v) * NCS + ccol] = c[v];
    }
}

// ---------------- edge-parallel message scatter ----------------
// thread = (edge, feature); F consecutive threads share one edge -> index/norm
// loads broadcast-coalesce; atomics land in the 12.8MB L2-resident accumulator.

template <int F>
__global__ void scatter_msgs(const float* __restrict__ H, const int* __restrict__ src,
                             const int* __restrict__ dst, const float* __restrict__ norm,
                             float* __restrict__ agg, int E) {
    int tid = blockIdx.x * blockDim.x + threadIdx.x;
    int e = tid / F;
    int f = tid - e * F;
    if (e >= E) return;
    int s = src[e], d = dst[e];
    atomicAdd(&agg[d * F + f], norm[e] * H[s * F + f]);
}

// ---------------- epilogue: self-loop + bias + ReLU ----------------
// self-loop contributes dinv[i]^2 * H[i] (w=1), folded here instead of N extra
// atomic edges.

template <int F>
__global__ void finish_layer(const float* __restrict__ agg, const float* __restrict__ H,
                             const float* __restrict__ dinv, const float* __restrict__ bias,
                             float* __restrict__ out, int Nn) {
    int tid = blockIdx.x * blockDim.x + threadIdx.x;
    int i = tid / F;
    int f = tid - i * F;
    if (i >= Nn) return;
    float di = dinv[i];
    float v = agg[tid] + di * di * H[tid] + bias[f];
    out[tid] = fmaxf(v, 0.0f);
}

// ---------------- final log-softmax over 40 classes ----------------

__global__ void log_softmax40(const float* __restrict__ h, float* __restrict__ out, int Nn) {
    int i = blockIdx.x * blockDim.x + threadIdx.x;
    if (i >= Nn) return;
    const float* r = h + i * OUTC;
    float m = -INFINITY;
#pragma unroll
    for (int f = 0; f < OUTC; ++f) m = fmaxf(m, r[f]);
    float s = 0.0f;
#pragma unroll
    for (int f = 0; f < OUTC; ++f) s += __expf(r[f] - m);
    float lse = m + __logf(s);
#pragma unroll
    for (int f = 0; f < OUTC; ++f) out[i * OUTC + f] = r[f] - lse;
}

// ---------------- host-side orchestration ----------------

extern "C" void kernel_launch(void* const* d_in, const int* in_sizes, int n_in,
                              void* d_out, int out_size, void* d_ws, size_t ws_size,
                              hipStream_t stream) {
    const float* x  = (const float*)d_in[0];
    const int*   ei = (const int*)d_in[1];
    const float* ea = (const float*)d_in[2];
    const float* W1 = (const float*)d_in[3];
    const float* b1 = (const float*)d_in[4];
    const float* W2 = (const float*)d_in[5];
    const float* b2 = (const float*)d_in[6];
    const float* W3 = (const float*)d_in[7];
    const float* b3 = (const float*)d_in[8];

    const int Nn = in_sizes[0] / IN_C;   // 50000
    const int E  = in_sizes[2];          // 1600000
    const int* src = ei;
    const int* dst = ei + E;

    float* ws   = (float*)d_ws;
    float* dinv = ws;  ws += Nn;                    // deg then dinv, in place
    float* norm = ws;  ws += E;
    float* H    = ws;  ws += (size_t)Nn * HIDC;     // GEMM output (pre-aggregation)
    float* agg  = ws;  ws += (size_t)Nn * HIDC;     // atomic accumulator
    float* hbuf = ws;  ws += (size_t)Nn * HIDC;     // layer activations (reused)
    float* w3p  = ws;  ws += HIDC * OUTP;           // padded W3 (64x48)

    const int B = 256;
    const int NH = Nn * HIDC;
    const int row_tiles = (Nn + 15) / 16;

    // --- normalization precompute (shared by all 3 layers) ---
    fill_f32<<<(Nn + B - 1) / B, B, 0, stream>>>(dinv, 1.0f, Nn);       // self-loop w=1
    deg_accum<<<(E + B - 1) / B, B, 0, stream>>>(dinv, dst, ea, E);
    rsqrt_inplace<<<(Nn + B - 1) / B, B, 0, stream>>>(dinv, Nn);
    compute_norm<<<(E + B - 1) / B, B, 0, stream>>>(norm, src, dst, ea, dinv, E);
    pad_w3<<<(HIDC * OUTP + B - 1) / B, B, 0, stream>>>(W3, w3p);

    // --- layer 1: 128 -> 64 ---
    {
        long long waves = (long long)row_tiles * (HIDC / 16);
        int blocks = (int)((waves * 32 + B - 1) / B);
        gemm_wmma_f32<IN_C, HIDC, HIDC><<<blocks, B, 0, stream>>>(x, W1, H, Nn);
        fill_f32<<<(NH + B - 1) / B, B, 0, stream>>>(agg, 0.0f, NH);
        long long tot = (long long)E * HIDC;
        scatter_msgs<HIDC><<<(int)((tot + B - 1) / B), B, 0, stream>>>(H, src, dst, norm, agg, E);
        finish_layer<HIDC><<<(NH + B - 1) / B, B, 0, stream>>>(agg, H, dinv, b1, hbuf, Nn);
    }

    // --- layer 2: 64 -> 64 ---
    {
        long long waves = (long long)row_tiles * (HIDC / 16);
        int blocks = (int)((waves * 32 + B - 1) / B);
        gemm_wmma_f32<HIDC, HIDC, HIDC><<<blocks, B, 0, stream>>>(hbuf, W2, H, Nn);
        fill_f32<<<(NH + B - 1) / B, B, 0, stream>>>(agg, 0.0f, NH);
        long long tot = (long long)E * HIDC;
        scatter_msgs<HIDC><<<(int)((tot + B - 1) / B), B, 0, stream>>>(H, src, dst, norm, agg, E);
        finish_layer<HIDC><<<(NH + B - 1) / B, B, 0, stream>>>(agg, H, dinv, b2, hbuf, Nn);
    }

    // --- layer 3: 64 -> 40 (B padded to 48 cols, stores clipped to 40) ---
    {
        long long waves = (long long)row_tiles * (OUTP / 16);
        int blocks = (int)((waves * 32 + B - 1) / B);
        gemm_wmma_f32<HIDC, OUTP, OUTC><<<blocks, B, 0, stream>>>(hbuf, w3p, H, Nn);
        const int NO = Nn * OUTC;
        fill_f32<<<(NO + B - 1) / B, B, 0, stream>>>(agg, 0.0f, NO);
        long long tot = (long long)E * OUTC;
        scatter_msgs<OUTC><<<(int)((tot + B - 1) / B), B, 0, stream>>>(H, src, dst, norm, agg, E);
        finish_layer<OUTC><<<(NO + B - 1) / B, B, 0, stream>>>(agg, H, dinv, b3, hbuf, Nn);
        log_softmax40<<<(Nn + B - 1) / B, B, 0, stream>>>(hbuf, (float*)d_out, Nn);
    }
}